// SelfAttentionLayerSparse_8100308320283
// MI455X (gfx1250) — compile-verified
//
#include <hip/hip_runtime.h>
#include <hip/hip_bf16.h>
#include <math.h>

typedef __attribute__((ext_vector_type(2))) float v2f;
typedef __attribute__((ext_vector_type(8))) float v8f;

#define N_NODES 20000
#define E_EDGES 640000
#define FIN     128
#define FQK     128
#define FV      128
#define FTOT    384
#define NHEAD   8
#define FH      16

// ---------------------------------------------------------------------------
// K0: zero output + expsum, init per-head max to -inf
// ---------------------------------------------------------------------------
__global__ __launch_bounds__(256) void init_buffers(float* __restrict__ out,
                                                    float* __restrict__ expsum,
                                                    float* __restrict__ hmax) {
    size_t i = (size_t)blockIdx.x * 256 + threadIdx.x;
    if (i < (size_t)N_NODES * FV) out[i] = 0.0f;
    if (i < (size_t)N_NODES * NHEAD) expsum[i] = 0.0f;
    if (i < NHEAD) hmax[i] = __int_as_float(0xff800000); // -inf
}

// ---------------------------------------------------------------------------
// K1: proj[N,384] = x[N,128] @ W[384,128]^T  via V_WMMA_F32_16X16X4_F32
// One wave per 16x16 output tile; K loop of 32 WMMA ops.
// A 16x4 f32 layout (ISA 7.12.2): lanes 0-15 hold M=lane, {VGPR0:K=0, VGPR1:K=1};
// lanes 16-31 hold M=lane-16, {VGPR0:K=2, VGPR1:K=3}.
// B 4x16 mirrors C-layout: VGPR0: K = 0 (lanes 0-15) / 2 (lanes 16-31), VGPR1: K=1/3,
// column N = lane&15; since B[k][n] = W[n][k], B addressing is symmetric with A.
// ---------------------------------------------------------------------------
__global__ __launch_bounds__(256) void gemm_proj(const float* __restrict__ x,
                                                 const float* __restrict__ W,
                                                 float* __restrict__ proj) {
    const int tid  = threadIdx.x;
    const int lane = tid & 31;
    const int wave = blockIdx.x * 8 + (tid >> 5);     // 3750 blocks * 8 waves = 30000 tiles
    const int ntiles = FTOT / 16;                     // 24
    const int mt = wave / ntiles;                     // 0..1249
    const int nt = wave - mt * ntiles;                // 0..23

    const int m0   = mt * 16;
    const int n0   = nt * 16;
    const int half = lane >> 4;                       // 0 or 1
    const int lr   = lane & 15;

    const float* xrow = x + (size_t)(m0 + lr) * FIN;  // A rows (M = lr)
    const float* wrow = W + (size_t)(n0 + lr) * FIN;  // B cols (N = lr)

    v8f c = {};
#pragma unroll
    for (int kk = 0; kk < FIN; kk += 4) {
        const int off = kk + 2 * half;                // even -> 8B aligned
        v2f a = *(const v2f*)(xrow + off);
        v2f b = *(const v2f*)(wrow + off);
        c = __builtin_amdgcn_wmma_f32_16x16x4_f32(
                /*neg_a=*/false, a, /*neg_b=*/false, b,
                /*c_mod=*/(short)0, c, /*reuse_a=*/false, /*reuse_b=*/false);
    }

    // C/D 16x16 f32 layout: VGPR i -> row (i + 8*half), col lr
#pragma unroll
    for (int i = 0; i < 8; ++i) {
        proj[(size_t)(m0 + i + 8 * half) * FTOT + n0 + lr] = c[i];
    }
}

// ---------------------------------------------------------------------------
// bit-punned float atomic max (addr must be initialized to -inf, never NaN)
// ---------------------------------------------------------------------------
__device__ __forceinline__ void atomicMaxF(float* addr, float v) {
    if (v >= 0.0f) atomicMax((int*)addr, __float_as_int(v));
    else           atomicMin((unsigned int*)addr, __float_as_uint(v));
}
__device__ __forceinline__ void atomicMaxF_shared(float* addr, float v) {
    if (v >= 0.0f) atomicMax((int*)addr, __float_as_int(v));
    else           atomicMin((unsigned int*)addr, __float_as_uint(v));
}

// ---------------------------------------------------------------------------
// K2: per-(edge,head) attention score + global per-head max
// aw[e,h] = 0.25 * dot(q[src[e],h,:], k[dest[e],h,:])  (16 dims, float4 loads)
// ---------------------------------------------------------------------------
__global__ __launch_bounds__(256) void edge_scores(const float* __restrict__ proj,
                                                   const int* __restrict__ src,
                                                   const int* __restrict__ dst,
                                                   float* __restrict__ aw,
                                                   float* __restrict__ hmax) {
    __shared__ float smax[NHEAD];
    const int tid = threadIdx.x;
    if (tid < NHEAD) smax[tid] = __int_as_float(0xff800000);
    __syncthreads();

    const size_t idx = (size_t)blockIdx.x * 256 + tid;   // grid sized exactly E*H
    const size_t e   = idx >> 3;
    const int    h   = (int)(idx & 7);
    const int    s   = src[e];
    const int    d   = dst[e];

    const float4* q4 = (const float4*)(proj + (size_t)s * FTOT + h * FH);
    const float4* k4 = (const float4*)(proj + (size_t)d * FTOT + FQK + h * FH);
    float acc = 0.0f;
#pragma unroll
    for (int j = 0; j < 4; ++j) {
        float4 q = q4[j], k = k4[j];
        acc += q.x * k.x + q.y * k.y + q.z * k.z + q.w * k.w;
    }
    const float val = acc * 0.25f;   // FH^-0.5 = 16^-0.5... = 0.25
    aw[idx] = val;

    atomicMaxF_shared(&smax[h], val);
    __syncthreads();
    if (tid < NHEAD) atomicMaxF(&hmax[tid], smax[tid]);
}

// ---------------------------------------------------------------------------
// K3: exp (global-max subtracted) + segment-sum over src
// ---------------------------------------------------------------------------
__global__ __launch_bounds__(256) void edge_exp(const int* __restrict__ src,
                                                const float* __restrict__ hmax,
                                                float* __restrict__ aw,       // in: scores, out: exp
                                                float* __restrict__ expsum) {
    const size_t idx = (size_t)blockIdx.x * 256 + threadIdx.x;  // exactly E*H threads
    const size_t e   = idx >> 3;
    const int    h   = (int)(idx & 7);
    const float  v   = expf(aw[idx] - hmax[h]) + 1e-8f;
    aw[idx] = v;
    atomicAdd(&expsum[(size_t)src[e] * NHEAD + h], v);
}

// ---------------------------------------------------------------------------
// K4: out[src] += (exp/expsum[src]) * v[dest] ; 128 threads per edge, coalesced
// ---------------------------------------------------------------------------
__global__ __launch_bounds__(256) void edge_accum(const float* __restrict__ proj,
                                                  const int* __restrict__ src,
                                                  const int* __restrict__ dst,
                                                  const float* __restrict__ expv,
                                                  const float* __restrict__ expsum,
                                                  float* __restrict__ out) {
    const int tid = threadIdx.x;
    const size_t e = (size_t)blockIdx.x * 2 + (tid >> 7);   // 2 edges / block
    const int c = tid & 127;                                // column in [0,128)
    const int h = c >> 4;
    const int s = src[e];
    const int d = dst[e];
    const float attn = expv[e * NHEAD + h] / expsum[(size_t)s * NHEAD + h];
    const float v = proj[(size_t)d * FTOT + 2 * FQK + c];
    atomicAdd(&out[(size_t)s * FV + c], attn * v);
}

// ---------------------------------------------------------------------------
extern "C" void kernel_launch(void* const* d_in, const int* in_sizes, int n_in,
                              void* d_out, int out_size, void* d_ws, size_t ws_size,
                              hipStream_t stream) {
    const float* x  = (const float*)d_in[0];   // [N,128]
    const float* W  = (const float*)d_in[1];   // [384,128]
    // d_in[2] = batch (unused by reference math)
    const int*   ei = (const int*)d_in[3];     // [2,E]
    const int* src  = ei;
    const int* dst  = ei + E_EDGES;
    float* out = (float*)d_out;

    // workspace layout (floats)
    float* proj   = (float*)d_ws;                                  // N*384
    float* aw     = proj   + (size_t)N_NODES * FTOT;               // E*H (scores -> exp)
    float* expsum = aw     + (size_t)E_EDGES * NHEAD;              // N*H
    float* hmax   = expsum + (size_t)N_NODES * NHEAD;              // H

    {
        size_t total = (size_t)N_NODES * FV;                       // 2,560,000
        init_buffers<<<(int)((total + 255) / 256), 256, 0, stream>>>(out, expsum, hmax);
    }
    gemm_proj  <<<3750,            256, 0, stream>>>(x, W, proj);        // 30000 tiles
    edge_scores<<<E_EDGES * NHEAD / 256, 256, 0, stream>>>(proj, src, dst, aw, hmax);
    edge_exp   <<<E_EDGES * NHEAD / 256, 256, 0, stream>>>(src, hmax, aw, expsum);
    edge_accum <<<E_EDGES / 2,     256, 0, stream>>>(proj, src, dst, aw, expsum, out);
}